// Preprocessor_65962107732374
// MI455X (gfx1250) — compile-verified
//
#include <hip/hip_runtime.h>
#include <stdint.h>

#define EPS        1e-5f
#define TILE       2048
#define PADZ       376          // z halo before tile start (even -> 8B aligned span)
#define ZLOAD      3072         // floats loaded per tile (6 async B64 rounds x 256 lanes)
#define ZSCAN      2799         // z values entering prefix scan
#define DLEN       2548         // d values (tile + 250 halo each side)
#define CHUNK      11           // per-thread serial scan chunk (coprime with 64 banks)
#define RED_BLOCKS 2048

// ---------------- pass 1: partial sums of x and x^2 ----------------
__global__ void reduce_pass1(const float* __restrict__ x, float* __restrict__ part, int n) {
    __shared__ float s1[256];
    __shared__ float s2[256];
    const int t = threadIdx.x;
    float a = 0.f, b = 0.f;
    const float4* x4 = (const float4*)x;
    const int n4 = n >> 2;
    for (int i = blockIdx.x * 256 + t; i < n4; i += gridDim.x * 256) {
        float4 v = x4[i];
        a += v.x + v.y + v.z + v.w;
        b += v.x * v.x + v.y * v.y + v.z * v.z + v.w * v.w;
    }
    if (blockIdx.x == 0) {                       // scalar tail (n not multiple of 4)
        for (int i = (n4 << 2) + t; i < n; i += 256) { float v = x[i]; a += v; b += v * v; }
    }
    s1[t] = a; s2[t] = b;
    __syncthreads();
    for (int s = 128; s > 0; s >>= 1) {
        if (t < s) { s1[t] += s1[t + s]; s2[t] += s2[t + s]; }
        __syncthreads();
    }
    if (t == 0) { part[2 * blockIdx.x] = s1[0]; part[2 * blockIdx.x + 1] = s2[0]; }
}

// ---------------- pass 2: final stats (double), zero output tail ----------------
__global__ void reduce_pass2(const float* __restrict__ part, float* __restrict__ stats,
                             float* __restrict__ out, int n, int out_size) {
    __shared__ double d1[256];
    __shared__ double d2[256];
    const int t = threadIdx.x;
    double a = 0.0, b = 0.0;
    for (int r = 0; r < RED_BLOCKS / 256; ++r) {
        int i = t + r * 256;
        a += (double)part[2 * i];
        b += (double)part[2 * i + 1];
    }
    d1[t] = a; d2[t] = b;
    __syncthreads();
    for (int s = 128; s > 0; s >>= 1) {
        if (t < s) { d1[t] += d1[t + s]; d2[t] += d2[t + s]; }
        __syncthreads();
    }
    if (t == 0) {
        double S = d1[0], Q = d2[0];
        double mean = S / (double)n;
        double var  = (Q - S * S / (double)n) / (double)(n - 1);   // ddof=1 (torch/jnp unbiased)
        stats[0] = (float)mean;
        stats[1] = (float)(1.0 / (sqrt(var) + (double)EPS));
    }
    // label (int64 zero) / any tail beyond N: zero bits
    for (int i = n + t; i < out_size; i += 256) out[i] = 0.f;
}

// generic (__shared__) pointer -> 32-bit LDS offset (low 32 bits of aperture address)
__device__ __forceinline__ uint32_t lds_off_of(const void* p) {
    return (uint32_t)(uintptr_t)p;
}

// ---------------- pass 3: tiled stencil via LDS prefix sums ----------------
__global__ void __launch_bounds__(256)
norm_main(const float* __restrict__ x, float* __restrict__ out,
          const float* __restrict__ stats, int n) {
    __shared__ __align__(16) float ZRAW[ZLOAD];  // raw x tile+halo
    __shared__ float CSZ[ZSCAN + 1];             // exclusive prefix of z
    __shared__ float D[DLEN];                    // residual d
    __shared__ float CSD[DLEN + 1];              // exclusive prefix of d^2
    __shared__ float TOT[256];

    const int t = threadIdx.x;
    const int tileStart = blockIdx.x * TILE;
    const float mean = stats[0];
    const float inv  = stats[1];
    const int zbase = tileStart - PADZ;

    const bool interior = (zbase >= 0) && (zbase + ZLOAD <= n);
    if (interior) {
        // CDNA5 async global->LDS copy: 6 rounds x 256 lanes x 8B = 12288 B
        const float* gsrc = x + zbase;                 // 8-byte aligned span
        const uint32_t ldsbase = lds_off_of(&ZRAW[0]);
        #pragma unroll
        for (int r = 0; r < 6; ++r) {
            int j = t + r * 256;                       // f64 element index
            uint32_t lo = ldsbase + (uint32_t)(j * 8);
            uint32_t go = (uint32_t)(j * 8);
            asm volatile("global_load_async_to_lds_b64 %0, %1, %2"
                         :: "v"(lo), "v"(go), "s"(gsrc) : "memory");
        }
        asm volatile("s_wait_asynccnt 0" ::: "memory");
    } else {
        // edge tiles: bounds-checked loads; OOB stores `mean` so z becomes exactly 0
        for (int r = 0; r < ZLOAD / 256; ++r) {
            int j = t + r * 256;
            int g = zbase + j;
            ZRAW[j] = (g >= 0 && g < n) ? x[g] : mean;
        }
    }
    __syncthreads();

    // ---- scan 1: exclusive prefix of z over ZSCAN values ----
    const int j0 = t * CHUNK;
    float run = 0.f;
    #pragma unroll
    for (int k = 0; k < CHUNK; ++k) {
        int j = j0 + k;
        if (j < ZSCAN) {
            float z = (ZRAW[j] - mean) * inv;
            run += z;
            CSZ[j + 1] = run;
        }
    }
    TOT[t] = run;
    __syncthreads();
    for (int s = 1; s < 256; s <<= 1) {
        float v = TOT[t];
        float a = (t >= s) ? TOT[t - s] : 0.f;
        __syncthreads();
        TOT[t] = v + a;
        __syncthreads();
    }
    {
        float off = (t > 0) ? TOT[t - 1] : 0.f;
        if (t == 0) CSZ[0] = 0.f;
        #pragma unroll
        for (int k = 0; k < CHUNK; ++k) {
            int j = j0 + k;
            if (j < ZSCAN) CSZ[j + 1] += off;
        }
    }
    __syncthreads();

    // ---- stage 2: d = z - ma251(z); scan d^2 ----
    float run2 = 0.f;
    #pragma unroll
    for (int k = 0; k < CHUNK; ++k) {
        int jd = j0 + k;
        if (jd < DLEN) {
            float z = (ZRAW[jd + 126] - mean) * inv;
            float d = z - (CSZ[jd + 252] - CSZ[jd + 1]) * (1.0f / 251.0f);
            int g = tileStart - 250 + jd;
            if (g < 0 || g >= n) d = 0.f;              // zero padding of the d signal
            D[jd] = d;
            run2 += d * d;
            CSD[jd + 1] = run2;
        }
    }
    TOT[t] = run2;
    __syncthreads();
    for (int s = 1; s < 256; s <<= 1) {
        float v = TOT[t];
        float a = (t >= s) ? TOT[t - s] : 0.f;
        __syncthreads();
        TOT[t] = v + a;
        __syncthreads();
    }
    {
        float off = (t > 0) ? TOT[t - 1] : 0.f;
        if (t == 0) CSD[0] = 0.f;
        #pragma unroll
        for (int k = 0; k < CHUNK; ++k) {
            int jd = j0 + k;
            if (jd < DLEN) CSD[jd + 1] += off;
        }
    }
    __syncthreads();

    // ---- stage 3: y = d / (sqrt(ma501(d^2)) + eps), coalesced stores ----
    #pragma unroll
    for (int r = 0; r < TILE / 256; ++r) {
        int i = t + r * 256;
        int gi = tileStart + i;
        if (gi < n) {
            int l = i + 250;
            float s501 = CSD[l + 251] - CSD[l - 250];
            float y = D[l] / (sqrtf(s501 * (1.0f / 501.0f)) + EPS);
            out[gi] = y;
        }
    }
}

extern "C" void kernel_launch(void* const* d_in, const int* in_sizes, int n_in,
                              void* d_out, int out_size, void* d_ws, size_t ws_size,
                              hipStream_t stream) {
    const float* x = (const float*)d_in[0];
    float* out = (float*)d_out;
    float* wsf = (float*)d_ws;
    const int n = in_sizes[0];

    float* partials = wsf;            // 2 * RED_BLOCKS floats
    float* stats    = wsf + 2 * RED_BLOCKS;  // {mean, 1/(std+eps)}

    hipLaunchKernelGGL(reduce_pass1, dim3(RED_BLOCKS), dim3(256), 0, stream, x, partials, n);
    hipLaunchKernelGGL(reduce_pass2, dim3(1), dim3(256), 0, stream, partials, stats, out, n, out_size);
    const int nblocks = (n + TILE - 1) / TILE;
    hipLaunchKernelGGL(norm_main, dim3(nblocks), dim3(256), 0, stream, x, out, stats, n);
}